// GCN_dgl_35021163331665
// MI455X (gfx1250) — compile-verified
//
#include <hip/hip_runtime.h>
#include <hip/hip_bf16.h>

typedef __attribute__((ext_vector_type(2))) float v2f;
typedef __attribute__((ext_vector_type(8))) float v8f;

#define BN_EPS 1e-5f
#define CH 128
// LDS K-stride (dwords) for transposed W: 132 => every lane of the wave hits a
// distinct LDS bank on the b64 B-fragment read, and (c*132+k)*4 is 8B-aligned.
#define WSTRIDE 132

__device__ __forceinline__ void fadd_atomic(float* p, float v) {
  unsafeAtomicAdd(p, v);  // native global_atomic_add_f32 (no CAS loop)
}

// ---------------- zero helpers ----------------
__global__ void zero_f4(float4* __restrict__ p, size_t n4) {
  size_t i = (size_t)blockIdx.x * blockDim.x + threadIdx.x;
  size_t stride = (size_t)gridDim.x * blockDim.x;
  float4 z = {0.f, 0.f, 0.f, 0.f};
  for (; i < n4; i += stride) p[i] = z;
}

__global__ void zero_f(float* __restrict__ p, int n) {
  int i = blockIdx.x * blockDim.x + threadIdx.x;
  if (i < n) p[i] = 0.f;
}

// ---------------- SpMM (128 ch): agg[dst] += ew * feat[src] ----------------
// One wave per edge; lane handles 4 contiguous channels: float4 gather
// (global_load_b128) + 4 native f32 atomics. feat (51.2MB) is L2-resident on
// MI455X (192MB L2) so the random gathers and scatter-atomics stay on-chip.
__global__ void spmm128_kernel(const float* __restrict__ feat,
                               const int* __restrict__ src,
                               const int* __restrict__ dst,
                               const float* __restrict__ ew,
                               float* __restrict__ agg, int E) {
  int e = blockIdx.x * 8 + (threadIdx.x >> 5);
  if (e >= E) return;
  int lane = threadIdx.x & 31;
  int s = src[e];
  int d = dst[e];
  float w = ew[e];
  const float4* fr = (const float4*)(feat + (size_t)s * CH);
  float4 v = fr[lane];
  float* o = agg + (size_t)d * CH + lane * 4;
  fadd_atomic(o + 0, v.x * w);
  fadd_atomic(o + 1, v.y * w);
  fadd_atomic(o + 2, v.z * w);
  fadd_atomic(o + 3, v.w * w);
}

// ---------------- SpMM (40 ch, layer 3 after GEMM-first reorder) ----------
// Rows are 40 floats; lanes 0..19 handle 2 channels each (float2 + 2 atomics).
__global__ void spmm40_kernel(const float* __restrict__ feat,
                              const int* __restrict__ src,
                              const int* __restrict__ dst,
                              const float* __restrict__ ew,
                              float* __restrict__ agg, int E) {
  int e = blockIdx.x * 8 + (threadIdx.x >> 5);
  if (e >= E) return;
  int lane = threadIdx.x & 31;
  if (lane >= 20) return;
  int s = src[e];
  int d = dst[e];
  float w = ew[e];
  float2 v = *(const float2*)(feat + (size_t)s * 40 + lane * 2);
  float* o = agg + (size_t)d * 40 + lane * 2;
  fadd_atomic(o + 0, v.x * w);
  fadd_atomic(o + 1, v.y * w);
}

// ---------------- GEMM via V_WMMA_F32_16X16X4_F32 ----------------
// out[N x Cout(ldo)] = A[N x 128] @ W[128 x Cout]. W staged TRANSPOSED in LDS
// (sWT[c*WSTRIDE + k]) so each B fragment is one conflict-free ds_load_b64.
// Block = 256 threads = 8 waves; each wave owns a 16-row stripe and NT 16x16
// accumulator tiles, stepping K by 4 per WMMA.
//
// ISA layouts (CDNA5 ISA):
//  A 16x4 f32 : lane l -> M = l&15 ; VGPR0/1 = K = (l>>4)*2 + {0,1}
//  B 4x16 f32 : lane l -> N = l&15 ; VGPR0/1 = K = (l>>4)*2 + {0,1}
//  C/D 16x16  : VGPR j -> M = j + (l>=16 ? 8 : 0), N = l&15
template <int NT>
__global__ void gemm_wmma_kernel(const float* __restrict__ A,
                                 const float* __restrict__ W,
                                 float* __restrict__ out,
                                 int Nrows, int Cout, int ldo) {
  constexpr int CP = NT * 16;
  __shared__ float sWT[CP * WSTRIDE];
  int tid = threadIdx.x;
  for (int idx = tid; idx < CH * CP; idx += 256) {
    int k = idx / CP;
    int c = idx % CP;
    sWT[c * WSTRIDE + k] = (c < Cout) ? W[k * Cout + c] : 0.f;
  }
  __syncthreads();

  int wave = tid >> 5;
  int lane = tid & 31;
  int row0 = blockIdx.x * 128 + wave * 16;
  if (row0 >= Nrows) return;  // N % 16 == 0: waves all-valid or all-invalid

  int m = lane & 15;           // M (A) / column-within-tile (B)
  int kh = (lane >> 4) * 2;    // K sub-offset: 0 or 2

  v8f acc[NT];
#pragma unroll
  for (int t = 0; t < NT; ++t) acc[t] = {};

  const float* Arow = A + (size_t)(row0 + m) * CH;

  for (int k0 = 0; k0 < CH; k0 += 4) {
    v2f a = *(const v2f*)(Arow + k0 + kh);
#pragma unroll
    for (int t = 0; t < NT; ++t) {
      v2f b = *(const v2f*)(&sWT[(t * 16 + m) * WSTRIDE + k0 + kh]);
      acc[t] = __builtin_amdgcn_wmma_f32_16x16x4_f32(
          /*neg_a=*/false, a, /*neg_b=*/false, b,
          /*c_mod=*/(short)0, acc[t], /*reuse_a=*/false, /*reuse_b=*/false);
    }
  }

  int rbase = row0 + (lane >> 4) * 8;
#pragma unroll
  for (int t = 0; t < NT; ++t) {
    int col = t * 16 + m;
    if (col < Cout) {
#pragma unroll
      for (int j = 0; j < 8; ++j) {
        out[(size_t)(rbase + j) * ldo + col] = acc[t][j];
      }
    }
  }
}

// ---------------- per-column sum / sumsq ----------------
// Stride is a multiple of 128 -> each thread's elements share one column, so
// partials accumulate in registers; one shared + one global atomic per thread.
__global__ void colstats_kernel(const float* __restrict__ g,
                                float* __restrict__ sums, int Nrows) {
  __shared__ float sh[256];
  sh[threadIdx.x] = 0.f;
  __syncthreads();
  size_t total = (size_t)Nrows * CH;
  size_t i = (size_t)blockIdx.x * 256 + threadIdx.x;
  size_t stride = (size_t)gridDim.x * 256;
  int col = (int)(i & (CH - 1));
  float s = 0.f, q = 0.f;
  for (; i < total; i += stride) {
    float v = g[i];
    s += v;
    q += v * v;
  }
  atomicAdd(&sh[col], s);
  atomicAdd(&sh[128 + col], q);
  __syncthreads();
  if (threadIdx.x < 128) {
    atomicAdd(&sums[threadIdx.x], sh[threadIdx.x]);
    atomicAdd(&sums[128 + threadIdx.x], sh[128 + threadIdx.x]);
  }
}

// ---------------- fused BN (biased var) + ReLU, in place ----------------
__global__ void bnrelu_kernel(float* __restrict__ g,
                              const float* __restrict__ sums,
                              const float* __restrict__ gamma,
                              const float* __restrict__ beta, int Nrows) {
  size_t total = (size_t)Nrows * CH;
  size_t i = (size_t)blockIdx.x * blockDim.x + threadIdx.x;
  size_t stride = (size_t)gridDim.x * blockDim.x;
  float invN = 1.0f / (float)Nrows;
  for (; i < total; i += stride) {
    int col = (int)(i & (CH - 1));
    float mean = sums[col] * invN;
    float var = sums[128 + col] * invN - mean * mean;
    float sc = gamma[col] * rsqrtf(var + BN_EPS);
    float v = (g[i] - mean) * sc + beta[col];
    g[i] = v > 0.f ? v : 0.f;
  }
}

// ---------------- layer-3 epilogue: out = agg40 + b3 ----------------
__global__ void biasout_kernel(const float* __restrict__ agg40,
                               const float* __restrict__ b3,
                               float* __restrict__ out, int total) {
  int i = blockIdx.x * blockDim.x + threadIdx.x;
  if (i < total) {
    int c = i % 40;
    out[i] = agg40[i] + b3[c];
  }
}

extern "C" void kernel_launch(void* const* d_in, const int* in_sizes, int n_in,
                              void* d_out, int out_size, void* d_ws, size_t ws_size,
                              hipStream_t stream) {
  const float* feat   = (const float*)d_in[0];
  const int*   src    = (const int*)d_in[1];
  const int*   dst    = (const int*)d_in[2];
  const float* ew     = (const float*)d_in[3];
  const float* W1     = (const float*)d_in[4];
  const float* W2     = (const float*)d_in[5];
  const float* W3     = (const float*)d_in[6];
  const float* b3     = (const float*)d_in[7];
  const float* gamma1 = (const float*)d_in[8];
  const float* beta1  = (const float*)d_in[9];
  const float* gamma2 = (const float*)d_in[10];
  const float* beta2  = (const float*)d_in[11];

  int N = in_sizes[0] / CH;  // 100000
  int E = in_sizes[1];       // 1600000

  float* agg  = (float*)d_ws;                // N*128 floats
  float* g    = agg + (size_t)N * CH;        // N*128 floats (h kept in place)
  float* sums = g + (size_t)N * CH;          // 256 floats: [sum | sumsq]
  float* t3   = agg;                         // layer-3: h2@W3, N*40 floats
  float* agg3 = agg + (size_t)N * 64;        // layer-3 aggregate, N*40 floats

  size_t n4 = (size_t)N * (CH / 4);
  int zgrid = 2048;
  int sgrid = (E + 7) / 8;
  int ggrid = (N + 127) / 128;
  int egrid = 2048;
  int bgrid = (N * 40 + 255) / 256;
  dim3 b256(256);

  // ---- layer 1 ----
  zero_f4<<<zgrid, b256, 0, stream>>>((float4*)agg, n4);
  spmm128_kernel<<<sgrid, b256, 0, stream>>>(feat, src, dst, ew, agg, E);
  gemm_wmma_kernel<8><<<ggrid, b256, 0, stream>>>(agg, W1, g, N, 128, 128);
  zero_f<<<1, b256, 0, stream>>>(sums, 256);
  colstats_kernel<<<egrid, b256, 0, stream>>>(g, sums, N);
  bnrelu_kernel<<<egrid, b256, 0, stream>>>(g, sums, gamma1, beta1, N);

  // ---- layer 2 ----
  zero_f4<<<zgrid, b256, 0, stream>>>((float4*)agg, n4);
  spmm128_kernel<<<sgrid, b256, 0, stream>>>(g, src, dst, ew, agg, E);
  gemm_wmma_kernel<8><<<ggrid, b256, 0, stream>>>(agg, W2, g, N, 128, 128);
  zero_f<<<1, b256, 0, stream>>>(sums, 256);
  colstats_kernel<<<egrid, b256, 0, stream>>>(g, sums, N);
  bnrelu_kernel<<<egrid, b256, 0, stream>>>(g, sums, gamma2, beta2, N);

  // ---- layer 3 (GEMM-first reorder: spmm(h@W3) == spmm(h)@W3) ----
  gemm_wmma_kernel<3><<<ggrid, b256, 0, stream>>>(g, W3, t3, N, 40, 40);
  zero_f4<<<zgrid, b256, 0, stream>>>((float4*)agg3, (size_t)N * 10);
  spmm40_kernel<<<sgrid, b256, 0, stream>>>(t3, src, dst, ew, agg3, E);
  biasout_kernel<<<bgrid, b256, 0, stream>>>(agg3, b3, (float*)d_out, N * 40);
}